// InfoNCELoss_21328807592556
// MI455X (gfx1250) — compile-verified
//
#include <hip/hip_runtime.h>
#include <hip/hip_bf16.h>
#include <math.h>
#include <stdint.h>

// ---------------------------------------------------------------------------
// Problem constants (match reference)
// ---------------------------------------------------------------------------
#define BN       8192      // batch
#define DN       1024      // embedding dim
#define INV_T    (1.0f / 0.07f)
#define MARGIN   0.5f
#define KHN      5         // hard negatives

typedef __attribute__((ext_vector_type(2))) float v2f;
typedef __attribute__((ext_vector_type(8))) float v8f;
typedef __attribute__((ext_vector_type(4))) int   v4i;

// ---------------------------------------------------------------------------
// CDNA5 async global->LDS copy path (ASYNCcnt), guarded so the file still
// compiles if the toolchain lacks the builtins.
// ROCm 7.2 signature (from diagnostics): (int4 as1* src, int4 as3* dst,
// imm offset, imm cpol).
// ---------------------------------------------------------------------------
#if defined(__has_builtin)
#  if __has_builtin(__builtin_amdgcn_global_load_async_to_lds_b128)
#    define USE_ASYNC_LDS 1
#  endif
#endif

#if defined(USE_ASYNC_LDS)
typedef __attribute__((address_space(1))) v4i* g1p4;
typedef __attribute__((address_space(3))) v4i* l3p4;

__device__ __forceinline__ void async_cp_b128(const float* gsrc, float* ldst) {
    __builtin_amdgcn_global_load_async_to_lds_b128(
        (g1p4)(uintptr_t)gsrc,
        (l3p4)(unsigned)(uintptr_t)ldst,
        /*imm offset*/0, /*cpol*/0);
}

#  if __has_builtin(__builtin_amdgcn_s_wait_asynccnt)
#    define WAIT_ASYNC(n) __builtin_amdgcn_s_wait_asynccnt(n)
#  else
#    define WAIT_ASYNC(n) asm volatile("s_wait_asynccnt %0" :: "i"(n) : "memory")
#  endif
#endif

// ---------------------------------------------------------------------------
// Kernel 1: inverse L2 norms for every row of t and g.
// grid.x = 2*BN blocks, 256 threads; each thread owns 4 consecutive floats.
// ---------------------------------------------------------------------------
__global__ void norm_kernel(const float* __restrict__ t,
                            const float* __restrict__ g,
                            float* __restrict__ rnt,
                            float* __restrict__ rng) {
    const int r   = blockIdx.x;
    const int tid = threadIdx.x;
    const float* src;
    float* dst;
    if (r < BN) { src = t + (size_t)r * DN;        dst = rnt + r; }
    else        { src = g + (size_t)(r - BN) * DN; dst = rng + (r - BN); }

    float4 v = *(const float4*)(src + tid * 4);
    float s = v.x * v.x + v.y * v.y + v.z * v.z + v.w * v.w;

    __shared__ float sm[256];
    sm[tid] = s;
    __syncthreads();
    #pragma unroll
    for (int off = 128; off > 0; off >>= 1) {
        if (tid < off) sm[tid] += sm[tid + off];
        __syncthreads();
    }
    if (tid == 0) *dst = 1.0f / fmaxf(sqrtf(sm[0]), 1e-12f);
}

// ---------------------------------------------------------------------------
// Kernel 2: sim = (t @ g^T) * invT * rnt[row] * rng[col]  via f32 WMMA.
//
// Block: 256 threads (8 waves), tile 128(M) x 128(N), K-chunk 32,
// double-buffered LDS panels filled by GLOBAL_LOAD_ASYNC_TO_LDS_B128 so the
// next panel's L2/HBM latency overlaps the current panel's WMMAs.
//
// f32 WMMA fragment layout (ISA 7.12.2):
//   A 16x4:  lane L<16 holds A[M=L][K=0..1] in v0..v1, lane L+16 holds K=2..3
//   B 4x16:  lane L<16 holds B[K=0..1][N=L],            lane L+16 holds K=2..3
//   -> both sides are row-major float2 loads for an NT GEMM.
//   C 16x16: VGPR r: lanes 0-15 -> (M=r, N=lane); lanes 16-31 -> (M=r+8).
// ---------------------------------------------------------------------------
#define MBLK 128
#define NBLK 128
#define KC   32
#define LSTR 36   // LDS row stride (dwords): 16B-aligned b128 stores,
                  // conflict-free b64 fragment reads (gcd(36,64)=4 pattern)

__global__ __launch_bounds__(256)
void gemm_wmma(const float* __restrict__ t,
               const float* __restrict__ g,
               const float* __restrict__ rnt,
               const float* __restrict__ rng,
               float* __restrict__ sim) {
    __shared__ float As[2][MBLK * LSTR];
    __shared__ float Bs[2][NBLK * LSTR];

    const int tid   = threadIdx.x;
    const int m_blk = blockIdx.y * MBLK;
    const int n_blk = blockIdx.x * NBLK;

    // staging mapping: thread -> (row, 16-col half)
    const int srow = tid >> 1;          // 0..127
    const int scol = (tid & 1) * 16;    // 0 or 16

    // wave / fragment mapping
    const int lane = tid & 31;
    const int wv   = tid >> 5;          // 0..7
    const int wm   = (wv >> 1) * 32;    // wave M offset within block: 0/32/64/96
    const int wn   = (wv & 1) * 64;     // wave N offset within block: 0/64
    const int lr   = lane & 15;         // M (A) / N (B) index on lanes
    const int kh   = (lane >> 4) * 2;   // K sub-offset: lo lanes K=0..1, hi K=2..3

    v8f acc[2][4];
    #pragma unroll
    for (int mt = 0; mt < 2; ++mt)
        #pragma unroll
        for (int nt = 0; nt < 4; ++nt)
            acc[mt][nt] = (v8f){0.f, 0.f, 0.f, 0.f, 0.f, 0.f, 0.f, 0.f};

    const float* agp = t + (size_t)(m_blk + srow) * DN + scol;
    const float* bgp = g + (size_t)(n_blk + srow) * DN + scol;

#if defined(USE_ASYNC_LDS)
    float* aspb[2] = { &As[0][srow * LSTR + scol], &As[1][srow * LSTR + scol] };
    float* bspb[2] = { &Bs[0][srow * LSTR + scol], &Bs[1][srow * LSTR + scol] };

    // Prologue: async-fill buffer 0 with chunk 0 (8 x b128 per thread).
    #pragma unroll
    for (int q = 0; q < 4; ++q) {
        async_cp_b128(agp + q * 4, aspb[0] + q * 4);
        async_cp_b128(bgp + q * 4, bspb[0] + q * 4);
    }

    for (int k0 = 0; k0 < DN; k0 += KC) {
        const int cur = (k0 / KC) & 1;
        if (k0 + KC < DN) {
            // Issue next panel into the other buffer, then drain only the
            // current panel's 8 copies (next 8 remain in flight).
            #pragma unroll
            for (int q = 0; q < 4; ++q) {
                async_cp_b128(agp + k0 + KC + q * 4, aspb[cur ^ 1] + q * 4);
                async_cp_b128(bgp + k0 + KC + q * 4, bspb[cur ^ 1] + q * 4);
            }
            WAIT_ASYNC(8);
        } else {
            WAIT_ASYNC(0);
        }
        __syncthreads();   // all waves' current-panel copies landed in LDS

        const float* Ac = As[cur];
        const float* Bc = Bs[cur];
        #pragma unroll
        for (int kk = 0; kk < KC; kk += 4) {
            v2f af[2], bf[4];
            #pragma unroll
            for (int mt = 0; mt < 2; ++mt)
                af[mt] = *(const v2f*)&Ac[(wm + mt * 16 + lr) * LSTR + kk + kh];
            #pragma unroll
            for (int nt = 0; nt < 4; ++nt)
                bf[nt] = *(const v2f*)&Bc[(wn + nt * 16 + lr) * LSTR + kk + kh];

            #pragma unroll
            for (int mt = 0; mt < 2; ++mt)
                #pragma unroll
                for (int nt = 0; nt < 4; ++nt)
                    acc[mt][nt] = __builtin_amdgcn_wmma_f32_16x16x4_f32(
                        false, af[mt], false, bf[nt],
                        (short)0, acc[mt][nt], false, false);
        }
        __syncthreads();   // done reading 'cur' before it is refilled
    }
#else
    // Fallback: synchronous staging through VGPRs, single buffer.
    float* asp = &As[0][srow * LSTR + scol];
    float* bsp = &Bs[0][srow * LSTR + scol];
    for (int k0 = 0; k0 < DN; k0 += KC) {
        __syncthreads();
        #pragma unroll
        for (int q = 0; q < 4; ++q) {
            float4 av = *(const float4*)(agp + k0 + q * 4);
            float4 bv = *(const float4*)(bgp + k0 + q * 4);
            *(float4*)(asp + q * 4) = av;
            *(float4*)(bsp + q * 4) = bv;
        }
        __syncthreads();
        #pragma unroll
        for (int kk = 0; kk < KC; kk += 4) {
            v2f af[2], bf[4];
            #pragma unroll
            for (int mt = 0; mt < 2; ++mt)
                af[mt] = *(const v2f*)&As[0][(wm + mt * 16 + lr) * LSTR + kk + kh];
            #pragma unroll
            for (int nt = 0; nt < 4; ++nt)
                bf[nt] = *(const v2f*)&Bs[0][(wn + nt * 16 + lr) * LSTR + kk + kh];
            #pragma unroll
            for (int mt = 0; mt < 2; ++mt)
                #pragma unroll
                for (int nt = 0; nt < 4; ++nt)
                    acc[mt][nt] = __builtin_amdgcn_wmma_f32_16x16x4_f32(
                        false, af[mt], false, bf[nt],
                        (short)0, acc[mt][nt], false, false);
        }
    }
#endif

    // Epilogue: fold 1/T and both row/col inverse norms, store sim.
    const int half = (lane >> 4) * 8;   // lanes 16-31 hold M = r + 8
    float rn_m[2][8];
    #pragma unroll
    for (int mt = 0; mt < 2; ++mt)
        #pragma unroll
        for (int r = 0; r < 8; ++r)
            rn_m[mt][r] = rnt[m_blk + wm + mt * 16 + r + half] * INV_T;
    float rg_n[4];
    #pragma unroll
    for (int nt = 0; nt < 4; ++nt)
        rg_n[nt] = rng[n_blk + wn + nt * 16 + lr];

    #pragma unroll
    for (int mt = 0; mt < 2; ++mt) {
        #pragma unroll
        for (int nt = 0; nt < 4; ++nt) {
            const int col = n_blk + wn + nt * 16 + lr;
            #pragma unroll
            for (int r = 0; r < 8; ++r) {
                const int row = m_blk + wm + mt * 16 + r + half;
                sim[(size_t)row * BN + col] = acc[mt][nt][r] * rn_m[mt][r] * rg_n[nt];
            }
        }
    }
}

// ---------------------------------------------------------------------------
// Kernel 3: per-row online logsumexp + diagonal + top-5 over label-mismatched
// columns + negative count. One block (256 threads) per row; b128 streaming.
// ---------------------------------------------------------------------------
__global__ __launch_bounds__(256)
void row_stats(const float* __restrict__ sim,
               const int* __restrict__ labels,
               float* __restrict__ row_inf,
               float* __restrict__ row_hard) {
    const int i   = blockIdx.x;
    const int tid = threadIdx.x;
    const float4* row4 = (const float4*)(sim + (size_t)i * BN);
    const int4*   lb4  = (const int4*)labels;
    const int li = labels[i];

    __shared__ float diag_s;
    __shared__ float sm_m[256];
    __shared__ float sm_s[256];
    __shared__ int   sm_c[256];
    __shared__ float sm_t[256 * KHN];

    float m = -INFINITY, s = 0.0f;
    float t5[KHN];
    #pragma unroll
    for (int k = 0; k < KHN; ++k) t5[k] = -INFINITY;
    int cnt = 0;

    for (int jb = tid; jb < BN / 4; jb += 256) {
        float4 v4 = row4[jb];
        int4   l4 = lb4[jb];
        const int j0 = jb * 4;
        float vv[4] = { v4.x, v4.y, v4.z, v4.w };
        int   ll[4] = { l4.x, l4.y, l4.z, l4.w };
        #pragma unroll
        for (int e = 0; e < 4; ++e) {
            float v = vv[e];
            if (j0 + e == i) diag_s = v;
            // online logsumexp over the full row
            if (v > m) { s = s * expf(m - v) + 1.0f; m = v; }
            else       { s += expf(v - m); }
            if (ll[e] != li) {
                cnt++;
                if (v > t5[KHN - 1]) {
                    t5[KHN - 1] = v;
                    #pragma unroll
                    for (int k = KHN - 1; k > 0; --k)
                        if (t5[k] > t5[k - 1]) { float tmp = t5[k]; t5[k] = t5[k - 1]; t5[k - 1] = tmp; }
                }
            }
        }
    }

    // log-depth merge (sorted-5 lists, online-lse combine, count sum)
    for (int off = 128; off > 0; off >>= 1) {
        sm_m[tid] = m; sm_s[tid] = s; sm_c[tid] = cnt;
        #pragma unroll
        for (int k = 0; k < KHN; ++k) sm_t[tid * KHN + k] = t5[k];
        __syncthreads();
        if (tid < off) {
            const int o = tid + off;
            float m2 = sm_m[o], s2 = sm_s[o];
            float nm = fmaxf(m, m2);
            s = s * expf(m - nm) + s2 * expf(m2 - nm);
            m = nm;
            cnt += sm_c[o];
            float ov[KHN], r5[KHN];
            #pragma unroll
            for (int k = 0; k < KHN; ++k) ov[k] = sm_t[o * KHN + k];
            int p = 0, q = 0;
            #pragma unroll
            for (int k = 0; k < KHN; ++k) {
                if (t5[p] >= ov[q]) r5[k] = t5[p++];
                else                r5[k] = ov[q++];
            }
            #pragma unroll
            for (int k = 0; k < KHN; ++k) t5[k] = r5[k];
        }
        __syncthreads();
    }

    if (tid == 0) {
        float lse = m + logf(s);
        row_inf[i] = lse - diag_s;            // -(log_softmax diagonal)
        int ntake = cnt < KHN ? cnt : KHN;
        float hl = 0.0f;
        #pragma unroll
        for (int k = 0; k < KHN; ++k)
            if (k < ntake) hl += fmaxf(t5[k] - MARGIN, 0.0f);
        row_hard[i] = (ntake > 0) ? hl / (float)ntake : 0.0f;
    }
}

// ---------------------------------------------------------------------------
// Kernel 4: reduce 8192 per-row values to the three output scalars.
// out[0] = total, out[1] = infonce, out[2] = hard_negative.
// ---------------------------------------------------------------------------
__global__ void final_reduce(const float* __restrict__ row_inf,
                             const float* __restrict__ row_hard,
                             float* __restrict__ out) {
    const int tid = threadIdx.x;
    float a = 0.0f, b = 0.0f;
    for (int j = tid; j < BN; j += 256) { a += row_inf[j]; b += row_hard[j]; }
    __shared__ float sa[256], sb[256];
    sa[tid] = a; sb[tid] = b;
    __syncthreads();
    #pragma unroll
    for (int off = 128; off > 0; off >>= 1) {
        if (tid < off) { sa[tid] += sa[tid + off]; sb[tid] += sb[tid + off]; }
        __syncthreads();
    }
    if (tid == 0) {
        float inf = sa[0] / (float)BN;
        float hrd = sb[0] / (float)BN;
        out[0] = inf + hrd;
        out[1] = inf;
        out[2] = hrd;
    }
}

// ---------------------------------------------------------------------------
// Launch
// ---------------------------------------------------------------------------
extern "C" void kernel_launch(void* const* d_in, const int* in_sizes, int n_in,
                              void* d_out, int out_size, void* d_ws, size_t ws_size,
                              hipStream_t stream) {
    const float* t      = (const float*)d_in[0];
    const float* g      = (const float*)d_in[1];
    const int*   labels = (const int*)d_in[2];
    float* out = (float*)d_out;
    float* ws  = (float*)d_ws;

    float* rnt = ws;                 // [BN]
    float* rng = ws + BN;            // [BN]
    float* ri  = ws + 2 * BN;        // [BN] per-row infonce
    float* rh  = ws + 3 * BN;        // [BN] per-row hard loss
    float* sim = out + 3;            // [BN*BN] output sim matrix

    norm_kernel<<<2 * BN, 256, 0, stream>>>(t, g, rnt, rng);

    dim3 grid(BN / NBLK, BN / MBLK);   // 64 x 64 blocks
    gemm_wmma<<<grid, 256, 0, stream>>>(t, g, rnt, rng, sim);

    row_stats<<<BN, 256, 0, stream>>>(sim, labels, ri, rh);

    final_reduce<<<1, 256, 0, stream>>>(ri, rh, out);
}